// SpatialLoss_67327907332131
// MI455X (gfx1250) — compile-verified
//
#include <hip/hip_runtime.h>

typedef float v2f __attribute__((ext_vector_type(2)));
typedef float v4f __attribute__((ext_vector_type(4)));
typedef float v8f __attribute__((ext_vector_type(8)));

#define MSE_BLOCKS  2048
#define MSE_THREADS 256

#define NSUB 100
#define PAD  112          // 7 * 16 (100 points padded to WMMA tiles)
#define NT_TILES 7
#define BT   128          // 4 wave32 waves

// ---------------------------------------------------------------------------
// Kernel A: streaming sum of squared differences. Bandwidth-bound: 64 MB read.
// Default temporal hints (RT) so the working set persists in the 192MB L2
// across graph replays. b128 loads, per-block LDS reduce, one partial/block.
// ---------------------------------------------------------------------------
__global__ __launch_bounds__(MSE_THREADS)
void mse_partial_kernel(const float* __restrict__ yp,
                        const float* __restrict__ yt,
                        float* __restrict__ partials, int n) {
    __shared__ float red[MSE_THREADS];
    const int tid = threadIdx.x;
    const long long gid    = (long long)blockIdx.x * MSE_THREADS + tid;
    const long long stride = (long long)gridDim.x * MSE_THREADS;
    const int n4 = n >> 2;

    const v4f* __restrict__ yp4 = (const v4f*)yp;
    const v4f* __restrict__ yt4 = (const v4f*)yt;

    float acc = 0.0f;
    for (long long i = gid; i < n4; i += stride) {
        v4f a = yp4[i];              // global_load_b128
        v4f b = yt4[i];
        v4f d = a - b;
        acc = fmaf(d.x, d.x, acc);
        acc = fmaf(d.y, d.y, acc);
        acc = fmaf(d.z, d.z, acc);
        acc = fmaf(d.w, d.w, acc);
    }
    // scalar tail (n % 4), handled once
    if (blockIdx.x == 0 && tid == 0) {
        for (int i = n4 << 2; i < n; ++i) {
            float d = yp[i] - yt[i];
            acc = fmaf(d, d, acc);
        }
    }

    red[tid] = acc;
    __syncthreads();
    #pragma unroll
    for (int s = MSE_THREADS / 2; s > 0; s >>= 1) {
        if (tid < s) red[tid] += red[tid + s];
        __syncthreads();
    }
    if (tid == 0) partials[blockIdx.x] = red[0];
}

// ---------------------------------------------------------------------------
// Kernel B: spatial penalty + final combine. One block, 4 wave32 waves.
// Gram matrix C*C^T via V_WMMA_F32_16X16X4_F32 (K=2 zero-padded to 4).
// ---------------------------------------------------------------------------
__global__ __launch_bounds__(BT)
void spatial_final_kernel(const float* __restrict__ yp,
                          const float* __restrict__ yt,
                          const float* __restrict__ coords,
                          const int*   __restrict__ perm,
                          const float* __restrict__ partials, int nPartials,
                          float* __restrict__ out, int n) {
    __shared__ float px[PAD], py[PAD], rr[PAD], n2[PAD];
    __shared__ float G[PAD * PAD];
    __shared__ float redMse[BT];
    __shared__ float redPen[BT];

    const int tid  = threadIdx.x;
    const int wave = tid >> 5;
    const int lane = tid & 31;

    // ---- gather sampled residuals / coordinates (zero-pad 100 -> 112) ----
    if (tid < PAD) {
        if (tid < NSUB) {
            int pi   = perm[tid];
            float x  = coords[2 * pi];
            float y  = coords[2 * pi + 1];
            px[tid]  = x;
            py[tid]  = y;
            n2[tid]  = fmaf(x, x, y * y);
            rr[tid]  = yp[pi] - yt[pi];
        } else {
            px[tid] = 0.0f; py[tid] = 0.0f; n2[tid] = 0.0f; rr[tid] = 0.0f;
        }
    }

    // ---- partial-sum the MSE block partials in parallel ----
    float msum = 0.0f;
    for (int i = tid; i < nPartials; i += BT) msum += partials[i];
    redMse[tid] = msum;
    __syncthreads();

    // ---- Gram matrix via WMMA: G = C (112x4, K padded) * C^T (4x112) ----
    // 49 16x16 output tiles distributed across the 4 waves; loop bounds are
    // wave-uniform so EXEC is all-ones at each WMMA (ISA requirement).
    for (int p = wave; p < NT_TILES * NT_TILES; p += 4) {
        const int tr = p / NT_TILES;
        const int tc = p % NT_TILES;
        v2f a, b;
        if (lane < 16) {
            // lanes 0-15: K=0 (x) in vgpr0, K=1 (y) in vgpr1; M/N = lane
            a.x = px[tr * 16 + lane];  a.y = py[tr * 16 + lane];
            b.x = px[tc * 16 + lane];  b.y = py[tc * 16 + lane];
        } else {
            // lanes 16-31 carry K=2,3 which are zero padding
            a = (v2f)(0.0f);
            b = (v2f)(0.0f);
        }
        v8f c = {};
        c = __builtin_amdgcn_wmma_f32_16x16x4_f32(
                /*neg_a=*/false, a, /*neg_b=*/false, b,
                /*c_mod=*/(short)0, c, /*reuse_a=*/false, /*reuse_b=*/false);

        const int ncol  = tc * 16 + (lane & 15);
        const int mbase = tr * 16 + (lane < 16 ? 0 : 8);
        #pragma unroll
        for (int v = 0; v < 8; ++v)
            G[(mbase + v) * PAD + ncol] = c[v];
    }
    __syncthreads();

    // ---- per-row 10-NN + penalty (row i handled by thread i) ----
    float pen = 0.0f;
    if (tid < NSUB) {
        const int i = tid;
        float bd[11];
        int   bi[11];
        #pragma unroll
        for (int k = 0; k < 11; ++k) { bd[k] = 3.4e38f; bi[k] = 0x7fffffff; }

        const float n2i = n2[i];
        for (int j = 0; j < NSUB; ++j) {
            float d2 = n2i + n2[j] - 2.0f * G[i * PAD + j];
            if (j == i) d2 = 0.0f;            // exact self-distance like ref
            d2 = fmaxf(d2, 0.0f);
            float d = sqrtf(d2);
            // keep 11 smallest; ties resolved to lower index (JAX top_k order)
            if (d < bd[10] || (d == bd[10] && j < bi[10])) {
                bd[10] = d; bi[10] = j;
                #pragma unroll
                for (int k = 10; k > 0; --k) {
                    bool sw = (bd[k] < bd[k - 1]) ||
                              (bd[k] == bd[k - 1] && bi[k] < bi[k - 1]);
                    if (sw) {
                        float td = bd[k]; bd[k] = bd[k - 1]; bd[k - 1] = td;
                        int   ti = bi[k]; bi[k] = bi[k - 1]; bi[k - 1] = ti;
                    }
                }
            }
        }
        // bd[1..10] = 10 nearest (self dropped); bd[10] = max of kept dists
        const float ri  = rr[i];
        const float inv = 1.0f / (bd[10] + 1e-8f);
        #pragma unroll
        for (int k = 1; k <= 10; ++k) {
            float nd = bd[k] * inv;
            float rd = fabsf(rr[bi[k]] - ri);
            pen += fabsf(rd - nd);
        }
        pen *= 0.1f;   // mean over K-1 = 10 neighbors
    }
    redPen[tid] = pen;
    __syncthreads();

    // ---- joint tree reduction of MSE partials and penalties ----
    #pragma unroll
    for (int s = BT / 2; s > 0; s >>= 1) {
        if (tid < s) {
            redMse[tid] += redMse[tid + s];
            redPen[tid] += redPen[tid + s];
        }
        __syncthreads();
    }
    if (tid == 0) {
        float mse     = redMse[0] / (float)n;
        float spatial = redPen[0] / (float)NSUB;
        out[0] = 0.8f * mse + 0.2f * spatial;
    }
}

// ---------------------------------------------------------------------------
extern "C" void kernel_launch(void* const* d_in, const int* in_sizes, int n_in,
                              void* d_out, int out_size, void* d_ws, size_t ws_size,
                              hipStream_t stream) {
    const float* yp     = (const float*)d_in[0];
    const float* yt     = (const float*)d_in[1];
    const float* coords = (const float*)d_in[2];
    const int*   perm   = (const int*)d_in[3];
    const int n = in_sizes[0];

    float* partials = (float*)d_ws;   // MSE_BLOCKS floats, fully rewritten each call

    mse_partial_kernel<<<MSE_BLOCKS, MSE_THREADS, 0, stream>>>(yp, yt, partials, n);
    spatial_final_kernel<<<1, BT, 0, stream>>>(yp, yt, coords, perm,
                                               partials, MSE_BLOCKS,
                                               (float*)d_out, n);
}